// GAN2L_65549790871886
// MI455X (gfx1250) — compile-verified
//
#include <hip/hip_runtime.h>
#include <hip/hip_bf16.h>

// ---------------------------------------------------------------------------
// GATv2 (2 layers, 8 heads) on N=201 nodes, dense reformulation.
// fp32 WMMA (V_WMMA_F32_16X16X4_F32) for all GEMMs.
// Round 2: branch-free WMMA inner loops (padded operands), 16x64 strips with
// 4 accumulator chains per wave, transposed B operand for the aggregation GEMM.
// ---------------------------------------------------------------------------

#define NN        201
#define NPAD      208          // 13 * 16
#define CH        1024
#define HEADS     8
#define CPH       128
#define MTILES    13
#define NEG_SLOPE 0.2f

typedef float v2f __attribute__((ext_vector_type(2)));
typedef float v8f __attribute__((ext_vector_type(8)));

__device__ inline v8f wmma4(v2f a, v2f b, v8f c) {
    return __builtin_amdgcn_wmma_f32_16x16x4_f32(
        false, a, false, b, (short)0, c, false, false);
}

// ---------------------------------------------------------------------------
// zero-fill scratch (cnt matrix + padded node buffers)
// ---------------------------------------------------------------------------
__global__ void zero_kernel(float* __restrict__ p, int n) {
    int i = blockIdx.x * blockDim.x + threadIdx.x;
    if (i < n) p[i] = 0.0f;
}

__global__ void copy_kernel(const float* __restrict__ src, float* __restrict__ dst, int n) {
    int i = blockIdx.x * blockDim.x + threadIdx.x;
    if (i < n) dst[i] = src[i];
}

// ---------------------------------------------------------------------------
// dense edge-count matrix (handles duplicate edges + self loops)
// ---------------------------------------------------------------------------
__global__ void count_kernel(const int* __restrict__ edges, int E,
                             unsigned* __restrict__ cnt) {
    int e = blockIdx.x * blockDim.x + threadIdx.x;
    if (e < E) {
        int s = edges[e];
        int d = edges[E + e];
        atomicAdd(&cnt[d * NPAD + s], 1u);
    } else if (e < E + NN) {
        int i = e - E;
        atomicAdd(&cnt[i * NPAD + i], 1u);
    }
}

// ---------------------------------------------------------------------------
// Node transform: Y[row, col] = sum_k X[row,k] * W[col,k] + bias[col]
// X is zero-padded [NPAD][K] -> NO guards in the K loop.
// One wave per 16x64 strip: 1 shared A fragment, 4 independent WMMA chains.
// ---------------------------------------------------------------------------
__global__ __launch_bounds__(256)
void gemm_node_wmma(const float* __restrict__ X,      // [NPAD][K], zero-padded
                    const float* __restrict__ W,      // [ncols x K] row-major
                    const float* __restrict__ bias,   // [ncols]
                    float* __restrict__ Y,            // [NPAD][ncols]
                    int K, int ncols, int n4tiles) {
    int wave = blockIdx.x * (blockDim.x >> 5) + (threadIdx.x >> 5);
    if (wave >= MTILES * n4tiles) return;
    int mt = wave % MTILES;
    int n4 = wave / MTILES;
    int lane = threadIdx.x & 31;
    int half = lane >> 4;            // 0: lanes 0-15 (K=0,1), 1: lanes 16-31 (K=2,3)
    int r    = lane & 15;
    const float* Xrow = X + (size_t)(mt * 16 + r) * K + (half << 1);
    int col0 = n4 * 64 + r;
    const float* Wp0 = W + (size_t)(col0)      * K + (half << 1);
    const float* Wp1 = W + (size_t)(col0 + 16) * K + (half << 1);
    const float* Wp2 = W + (size_t)(col0 + 32) * K + (half << 1);
    const float* Wp3 = W + (size_t)(col0 + 48) * K + (half << 1);
    v8f acc0 = {}, acc1 = {}, acc2 = {}, acc3 = {};
    #pragma unroll 2
    for (int k = 0; k < K; k += 4) {
        v2f a  = *(const v2f*)(Xrow + k);
        v2f b0 = *(const v2f*)(Wp0 + k);
        v2f b1 = *(const v2f*)(Wp1 + k);
        v2f b2 = *(const v2f*)(Wp2 + k);
        v2f b3 = *(const v2f*)(Wp3 + k);
        acc0 = wmma4(a, b0, acc0);
        acc1 = wmma4(a, b1, acc1);
        acc2 = wmma4(a, b2, acc2);
        acc3 = wmma4(a, b3, acc3);
    }
    // C layout: VGPR i -> M = i (lanes 0-15) / i+8 (lanes 16-31), N = lane&15
    #define STORE_STRIP(ACC, SUB)                                             \
        {                                                                     \
            int col = col0 + (SUB) * 16;                                      \
            float bv = bias[col];                                             \
            _Pragma("unroll")                                                 \
            for (int i = 0; i < 8; i++) {                                     \
                int orow = mt * 16 + half * 8 + i;                            \
                if (orow < NN) Y[(size_t)orow * ncols + col] = ACC[i] + bv;   \
            }                                                                 \
        }
    STORE_STRIP(acc0, 0)
    STORE_STRIP(acc1, 1)
    STORE_STRIP(acc2, 2)
    STORE_STRIP(acc3, 3)
    #undef STORE_STRIP
}

// ---------------------------------------------------------------------------
// LDS-tiled transpose: XT[c][n] = X[n][c];  X:[NPAD][CH] -> XT:[CH][NPAD]
// ---------------------------------------------------------------------------
__global__ __launch_bounds__(256)
void transpose_kernel(const float* __restrict__ X, float* __restrict__ XT) {
    __shared__ float tile[16][17];
    int tn = blockIdx.x % (NPAD / 16);
    int tc = blockIdx.x / (NPAD / 16);
    int tx = threadIdx.x & 15;
    int ty = threadIdx.x >> 4;
    tile[ty][tx] = X[(size_t)(tn * 16 + ty) * CH + tc * 16 + tx];   // coalesced read
    __syncthreads();
    XT[(size_t)(tc * 16 + ty) * NPAD + tn * 16 + tx] = tile[tx][ty]; // coalesced write
}

// ---------------------------------------------------------------------------
// GATv2 scores: S[dst][src][h] = sum_c att[h,c] * leakyrelu(xl[src,hc]+xr[dst,hc])
// One wave per (dst,src) pair; skipped if no edge (wave-uniform branch).
// ---------------------------------------------------------------------------
__global__ __launch_bounds__(256)
void score_kernel(const float* __restrict__ xl, const float* __restrict__ xr,
                  const float* __restrict__ att, const unsigned* __restrict__ cnt,
                  float* __restrict__ S) {
    int wave = blockIdx.x * (blockDim.x >> 5) + (threadIdx.x >> 5);
    if (wave >= NN * NN) return;
    int dst = wave / NN;
    int src = wave % NN;
    if (cnt[dst * NPAD + src] == 0u) return;
    int lane = threadIdx.x & 31;
    const float* pl = xl + (size_t)src * CH;
    const float* pr = xr + (size_t)dst * CH;
    float acc[HEADS];
    #pragma unroll
    for (int h = 0; h < HEADS; h++) {
        float a = 0.0f;
        #pragma unroll
        for (int j = 0; j < 4; j++) {              // 128 ch / 32 lanes
            int hc = h * CPH + j * 32 + lane;
            float v = pl[hc] + pr[hc];
            v = (v > 0.0f) ? v : NEG_SLOPE * v;
            a += att[hc] * v;
        }
        #pragma unroll
        for (int off = 16; off; off >>= 1) a += __shfl_xor(a, off, 32);
        acc[h] = a;
    }
    if (lane == 0) {
        float* out = S + ((size_t)dst * NPAD + src) * HEADS;
        #pragma unroll
        for (int h = 0; h < HEADS; h++) out[h] = acc[h];
    }
}

// ---------------------------------------------------------------------------
// Segment softmax over src (count-weighted for duplicate edges).
// One wave per (dst, head). Writes dense P[h][dst][src] incl. exact-zero pads.
// ---------------------------------------------------------------------------
__global__ __launch_bounds__(256)
void softmax_kernel(const float* __restrict__ S, const unsigned* __restrict__ cnt,
                    float* __restrict__ P) {
    int wave = blockIdx.x * (blockDim.x >> 5) + (threadIdx.x >> 5);
    if (wave >= NN * HEADS) return;
    int dst = wave >> 3;
    int h   = wave & 7;
    int lane = threadIdx.x & 31;
    float m = -3.4e38f;
    for (int s0 = 0; s0 < NPAD; s0 += 32) {
        int s = s0 + lane;
        unsigned c = (s < NN) ? cnt[dst * NPAD + s] : 0u;
        float sc = c ? S[((size_t)dst * NPAD + s) * HEADS + h] : -3.4e38f;
        m = fmaxf(m, sc);
    }
    #pragma unroll
    for (int off = 16; off; off >>= 1) m = fmaxf(m, __shfl_xor(m, off, 32));
    float dsum = 0.0f;
    for (int s0 = 0; s0 < NPAD; s0 += 32) {
        int s = s0 + lane;
        unsigned c = (s < NN) ? cnt[dst * NPAD + s] : 0u;
        float ex = c ? (float)c * __expf(S[((size_t)dst * NPAD + s) * HEADS + h] - m)
                     : 0.0f;
        dsum += ex;
    }
    #pragma unroll
    for (int off = 16; off; off >>= 1) dsum += __shfl_xor(dsum, off, 32);
    float inv = 1.0f / (dsum + 1e-16f);
    float* Ph = P + (size_t)h * NPAD * NPAD + (size_t)dst * NPAD;
    for (int s0 = 0; s0 < NPAD; s0 += 32) {
        int s = s0 + lane;
        unsigned c = (s < NN) ? cnt[dst * NPAD + s] : 0u;
        float ex = c ? (float)c * __expf(S[((size_t)dst * NPAD + s) * HEADS + h] - m) * inv
                     : 0.0f;
        Ph[s] = ex;
    }
}

// ---------------------------------------------------------------------------
// Aggregation: Y[dst, h*128+c] = sum_src P[h][dst][src] * XLT[h*128+c][src] + b
// 8 dense [201x201]x[201x128] GEMMs via fp32 WMMA. B operand pre-transposed so
// every fragment load is a contiguous b64. 16x64 strip per wave, 4 acc chains.
// ---------------------------------------------------------------------------
__global__ __launch_bounds__(256)
void agg_wmma(const float* __restrict__ P,    // [8][NPAD][NPAD]
              const float* __restrict__ XLT,  // [CH][NPAD]  (xl transposed)
              const float* __restrict__ bias, // [CH]
              float* __restrict__ Y,          // [NPAD][CH]
              int relu) {
    int wave = blockIdx.x * (blockDim.x >> 5) + (threadIdx.x >> 5);
    const int total = HEADS * MTILES * 2;     // 2 strips of 64 cols per head
    if (wave >= total) return;
    int h   = wave / (MTILES * 2);
    int rem = wave % (MTILES * 2);
    int mt  = rem % MTILES;
    int q   = rem / MTILES;                   // 0 or 1: which 64-col strip
    int lane = threadIdx.x & 31;
    int half = lane >> 4;
    int r    = lane & 15;
    const float* Prow = P + (size_t)h * NPAD * NPAD
                          + (size_t)(mt * 16 + r) * NPAD + (half << 1);
    int col0 = h * CPH + q * 64 + r;
    const float* B0 = XLT + (size_t)(col0)      * NPAD + (half << 1);
    const float* B1 = XLT + (size_t)(col0 + 16) * NPAD + (half << 1);
    const float* B2 = XLT + (size_t)(col0 + 32) * NPAD + (half << 1);
    const float* B3 = XLT + (size_t)(col0 + 48) * NPAD + (half << 1);
    v8f acc0 = {}, acc1 = {}, acc2 = {}, acc3 = {};
    #pragma unroll 2
    for (int k = 0; k < NPAD; k += 4) {
        v2f a  = *(const v2f*)(Prow + k);
        v2f b0 = *(const v2f*)(B0 + k);
        v2f b1 = *(const v2f*)(B1 + k);
        v2f b2 = *(const v2f*)(B2 + k);
        v2f b3 = *(const v2f*)(B3 + k);
        acc0 = wmma4(a, b0, acc0);
        acc1 = wmma4(a, b1, acc1);
        acc2 = wmma4(a, b2, acc2);
        acc3 = wmma4(a, b3, acc3);
    }
    #define STORE_STRIP(ACC, SUB)                                             \
        {                                                                     \
            int col = col0 + (SUB) * 16;                                      \
            float bv = bias[col];                                             \
            _Pragma("unroll")                                                 \
            for (int i = 0; i < 8; i++) {                                     \
                int orow = mt * 16 + half * 8 + i;                            \
                if (orow < NN) {                                              \
                    float v = ACC[i] + bv;                                    \
                    if (relu) v = (v > 0.0f) ? v : 0.0f;                      \
                    Y[(size_t)orow * CH + col] = v;                           \
                }                                                             \
            }                                                                 \
        }
    STORE_STRIP(acc0, 0)
    STORE_STRIP(acc1, 1)
    STORE_STRIP(acc2, 2)
    STORE_STRIP(acc3, 3)
    #undef STORE_STRIP
}

// ---------------------------------------------------------------------------
// fuse_average pooling + 2-class linear classifier (single block)
// ---------------------------------------------------------------------------
__global__ __launch_bounds__(256)
void pool_clf_kernel(const float* __restrict__ H,
                     const float* __restrict__ clfW,
                     const float* __restrict__ clfb,
                     float* __restrict__ out) {
    __shared__ float red0[256];
    __shared__ float red1[256];
    int t = threadIdx.x;
    float p0 = 0.0f, p1 = 0.0f;
    for (int k = t; k < CH; k += 256) {
        float s1 = 0.0f, s2 = 0.0f;
        for (int n = 0; n < 100; n++) {
            s1 += H[(size_t)n * CH + k];
            s2 += H[(size_t)(n + 100) * CH + k];
        }
        float avg = (s1 * 0.01f + s2 * 0.01f + H[(size_t)200 * CH + k]) * (1.0f / 3.0f);
        p0 += avg * clfW[k];
        p1 += avg * clfW[CH + k];
    }
    red0[t] = p0; red1[t] = p1;
    __syncthreads();
    for (int s = 128; s; s >>= 1) {
        if (t < s) { red0[t] += red0[t + s]; red1[t] += red1[t + s]; }
        __syncthreads();
    }
    if (t == 0) {
        out[0] = red0[0] + clfb[0];
        out[1] = red1[0] + clfb[1];
    }
}

// ---------------------------------------------------------------------------
// launch
// ---------------------------------------------------------------------------
extern "C" void kernel_launch(void* const* d_in, const int* in_sizes, int n_in,
                              void* d_out, int out_size, void* d_ws, size_t ws_size,
                              hipStream_t stream) {
    const float* x     = (const float*)d_in[0];
    const int*   edges = (const int*)  d_in[1];
    const float* W1l   = (const float*)d_in[2];
    const float* b1l   = (const float*)d_in[3];
    const float* W1r   = (const float*)d_in[4];
    const float* b1r   = (const float*)d_in[5];
    const float* att1  = (const float*)d_in[6];
    const float* bias1 = (const float*)d_in[7];
    const float* W2l   = (const float*)d_in[8];
    const float* b2l   = (const float*)d_in[9];
    const float* W2r   = (const float*)d_in[10];
    const float* b2r   = (const float*)d_in[11];
    const float* att2  = (const float*)d_in[12];
    const float* bias2 = (const float*)d_in[13];
    const float* clfW  = (const float*)d_in[14];
    const float* clfb  = (const float*)d_in[15];
    float* out = (float*)d_out;
    const int E = in_sizes[1] / 2;

    // workspace layout (bytes)
    char* ws = (char*)d_ws;
    unsigned* cnt = (unsigned*)(ws);                // NPAD*NPAD u32      @ 0
    float* xpad = (float*)(ws + 173056);            // NPAD*CH
    float* xl   = (float*)(ws + 1025024);           // NPAD*CH
    float* xr   = (float*)(ws + 1876992);           // NPAD*CH
    float* h1   = (float*)(ws + 2728960);           // NPAD*CH
    float* h2   = (float*)(ws + 3580928);           // NPAD*CH
    float* XLT  = (float*)(ws + 4432896);           // CH*NPAD (fully rewritten)
    float* S    = (float*)(ws + 5284864);           // NPAD*NPAD*HEADS
    float* P    = (float*)(ws + 6669312);           // HEADS*NPAD*NPAD

    // zero cnt + padded node buffers every launch (deterministic)
    {
        int nz = (173056 + 5 * NPAD * CH * 4) / 4;
        zero_kernel<<<(nz + 255) / 256, 256, 0, stream>>>((float*)ws, nz);
    }
    copy_kernel<<<(NN * CH + 255) / 256, 256, 0, stream>>>(x, xpad, NN * CH);
    {
        int etot = E + NN;
        count_kernel<<<(etot + 255) / 256, 256, 0, stream>>>(edges, E, cnt);
    }

    const int gemmBlk  = (MTILES * 16 + 7) / 8;          // 208 waves -> 26 blocks
    const int scoreBlk = (NN * NN + 7) / 8;              // 5051
    const int smaxBlk  = (NN * HEADS + 7) / 8;           // 201
    const int aggBlk   = (HEADS * MTILES * 2 + 7) / 8;   // 26
    const int trBlk    = (NPAD / 16) * (CH / 16);        // 832

    // ---- layer 1 ----
    gemm_node_wmma<<<gemmBlk, 256, 0, stream>>>(xpad, W1l, b1l, xl, CH, CH, 16);
    gemm_node_wmma<<<gemmBlk, 256, 0, stream>>>(xpad, W1r, b1r, xr, CH, CH, 16);
    score_kernel  <<<scoreBlk, 256, 0, stream>>>(xl, xr, att1, cnt, S);
    softmax_kernel<<<smaxBlk, 256, 0, stream>>>(S, cnt, P);
    transpose_kernel<<<trBlk, 256, 0, stream>>>(xl, XLT);
    agg_wmma      <<<aggBlk, 256, 0, stream>>>(P, XLT, bias1, h1, /*relu=*/1);

    // ---- layer 2 (reuse xl/xr buffers; h1 already padded+zeroed) ----
    gemm_node_wmma<<<gemmBlk, 256, 0, stream>>>(h1, W2l, b2l, xl, CH, CH, 16);
    gemm_node_wmma<<<gemmBlk, 256, 0, stream>>>(h1, W2r, b2r, xr, CH, CH, 16);
    score_kernel  <<<scoreBlk, 256, 0, stream>>>(xl, xr, att2, cnt, S);
    softmax_kernel<<<smaxBlk, 256, 0, stream>>>(S, cnt, P);
    transpose_kernel<<<trBlk, 256, 0, stream>>>(xl, XLT);
    agg_wmma      <<<aggBlk, 256, 0, stream>>>(P, XLT, bias2, h2, /*relu=*/0);

    // ---- pooling + classifier ----
    pool_clf_kernel<<<1, 256, 0, stream>>>(h2, clfW, clfb, out);
}